// DeepMultiPlasticNet_61607010894102
// MI455X (gfx1250) — compile-verified
//
#include <hip/hip_runtime.h>
#include <hip/hip_bf16.h>
#include <math.h>

// ---------------------------------------------------------------------------
// DeepMultiPlasticNet for MI455X (gfx1250, wave32).
//
// Strategy (see analysis): total FLOPs ~1.5G -> latency/LDS-bound, not compute.
// 1) v = x @ W_in^T + b_in for all T*B rows at once: fp32 WMMA GEMM
//    (V_WMMA_F32_16X16X4_F32), 3072x384, K=64.  Time-parallel, hoisted out of
//    the scan.
// 2) The T=48 sequential scan: one workgroup per batch element b keeps the 15
//    live strictly-upper 64x64 blocks of M[b] resident in LDS (240 KB of the
//    320 KB/WGP) for the whole kernel.  M starts at zero and never touches
//    global memory -> eliminates ~3.6 GB of M traffic entirely.
//    Diagonal-block entries of M are never read by the recurrence and do not
//    affect the output, so they are not materialized.
// 3) logits = h_last @ W_out^T + b_out : 64x16(pad)x384 fp32 WMMA GEMM.
// ---------------------------------------------------------------------------

typedef __attribute__((ext_vector_type(2))) float v2f;
typedef __attribute__((ext_vector_type(8))) float v8f;

#define T_STEPS 48
#define BSZ     64
#define D_IN    64
#define D_H     384
#define D_OUT   10
#define NBLK    6              // 384 / 64
#define NUPPER  15             // strictly-upper 64x64 blocks
#define BLKELEM 4096           // 64*64

// lambda = exp(-1/100), eta = (1-lambda)*0.1, C = eta / sqrt(H)
#define LAM_F   0.99004983374916811f
#define ETA_F   0.00099501662508319f
#define ISQH_F  0.051031036307982884f

// ---------------------------------------------------------------------------
// Kernel 1: v[(t*B+b), h] = sum_k x[t,b,k] * W_in[h,k] + b_in[h]
// GEMM (3072 x 384, K=64) via V_WMMA_F32_16X16X4_F32, one 16x16 tile / wave32.
// A-frag (16x4 f32, 2 VGPRs): lane&15 = M row; half=lane>>4; VGPR v holds
//   K = 2*half + v.   B-frag (4x16) mirrored: lane&15 = N col, same K mapping.
// C/D (16x16 f32, 8 VGPRs): VGPR r -> row r + 8*half, col = lane&15.
// ---------------------------------------------------------------------------
__global__ __launch_bounds__(128)
void v_gemm_wmma(const float* __restrict__ x,      // (T,B,64) row-major
                 const float* __restrict__ W_in,   // (384,64) row-major
                 const float* __restrict__ b_in,   // (384)
                 float* __restrict__ v_out)        // (T*B,384)
{
    const int lane = threadIdx.x & 31;
    const int wave = threadIdx.x >> 5;
    const int tile = blockIdx.x * 4 + wave;        // 192*24 = 4608 tiles
    const int rt   = tile / 24;                    // row tile (T*B / 16)
    const int ct   = tile % 24;                    // col tile (384 / 16)
    const int r0   = rt * 16;
    const int c0   = ct * 16;
    const int half = lane >> 4;
    const int l15  = lane & 15;

    v8f acc = {};
    const float* arow = x    + (r0 + l15) * 64;    // A row  (M = l15)
    const float* bcol = W_in + (c0 + l15) * 64;    // B(k,n) = W_in[n,k]
    #pragma unroll 4
    for (int kk = 0; kk < 64; kk += 4) {
        const int ka = kk + 2 * half;
        v2f a, b;
        a.x = arow[ka];  a.y = arow[ka + 1];
        b.x = bcol[ka];  b.y = bcol[ka + 1];
        acc = __builtin_amdgcn_wmma_f32_16x16x4_f32(
                  false, a, false, b, (short)0, acc, false, false);
    }
    const int   n    = c0 + l15;
    const float bias = b_in[n];
    #pragma unroll
    for (int vr = 0; vr < 8; ++vr) {
        const int m = r0 + vr + 8 * half;
        v_out[m * D_H + n] = acc[vr] + bias;
    }
}

// ---------------------------------------------------------------------------
// Kernel 2: the sequential plastic scan.  grid = 64 (one WG per batch elem),
// block = 256 threads (8 wave32s).  Dynamic LDS:
//   Mlds[15*4096]  strictly-upper blocks of M[b], fp32, resident all 48 steps
//   hbuf[384], vbuf[384], part[256]
// Block storage order: (bi,bj) for bj=1..5, bi=0..bj-1  -> flat blk index.
// ---------------------------------------------------------------------------
__global__ __launch_bounds__(256, 1)
void plastic_scan(const float* __restrict__ v_all,  // (T*B,384)
                  const float* __restrict__ W,      // (384,384)
                  float* __restrict__ h_out)        // (B,384) final h
{
    extern __shared__ float smem[];
    float* Mlds = smem;                       // 61440 f32 = 240 KB
    float* hbuf = Mlds + NUPPER * BLKELEM;    // 384
    float* vbuf = hbuf + D_H;                 // 384
    float* part = vbuf + D_H;                 // 256

    const int b   = blockIdx.x;
    const int tid = threadIdx.x;
    const int col = tid & 63;
    const int q   = tid >> 6;                 // quarter 0..3 (K-split)

    // M[b] starts at zero.
    for (int e = tid; e < NUPPER * BLKELEM; e += 256) Mlds[e] = 0.0f;
    __syncthreads();

    // nibble-packed (bi,bj) tables for flat block index 0..14
    const unsigned long long BI_PACK = 0x432103210210100ull;
    const unsigned long long BJ_PACK = 0x555554444333221ull;
    const float Cupd = ETA_F * ISQH_F;

    for (int t = 0; t < T_STEPS; ++t) {
        // stage v_t[b,:] (hits L2; tiny)
        const float* vrow = v_all + (t * BSZ + b) * D_H;
        for (int i = tid; i < D_H; i += 256) vbuf[i] = vrow[i];
        __syncthreads();

        // block 0: no recurrence
        if (tid < 64) hbuf[tid] = tanhf(vbuf[tid]);
        __syncthreads();

        // blocks 1..5: h_j = tanh(v_j + h_{<j} @ ((1+M)∘W)[:,j])
        for (int bj = 1; bj < NBLK; ++bj) {
            const int    j0   = bj * 64;
            const float* Mcol = Mlds + (bj * (bj - 1) / 2) * BLKELEM;
            float acc = 0.0f;
            for (int i = q; i < j0; i += 4) {
                const float hi = hbuf[i];                              // LDS bcast
                const float m  = Mcol[(i >> 6) * BLKELEM + (i & 63) * 64 + col];
                const float w  = W[i * D_H + j0 + col];                // L2/WGP$
                acc += hi * (1.0f + m) * w;
            }
            part[tid] = acc;
            __syncthreads();
            if (q == 0) {
                const float s = part[col] + part[64 + col] +
                                part[128 + col] + part[192 + col] +
                                vbuf[j0 + col];
                hbuf[j0 + col] = tanhf(s);
            }
            __syncthreads();
        }

        // plasticity update: M_ij = LAM*M_ij + C * h_i h_j^T (upper blocks)
        for (int blk = 0; blk < NUPPER; ++blk) {
            const int bi = (int)((BI_PACK >> (4 * blk)) & 15ull);
            const int bj = (int)((BJ_PACK >> (4 * blk)) & 15ull);
            float* Mb = Mlds + blk * BLKELEM;
            const float* hr = hbuf + bi * 64;
            const float* hc = hbuf + bj * 64;
            for (int e = tid; e < BLKELEM; e += 256) {
                const int r = e >> 6, c = e & 63;
                Mb[e] = LAM_F * Mb[e] + Cupd * hr[r] * hc[c];
            }
        }
        __syncthreads();   // M update done before next step overwrites hbuf
    }

    for (int i = tid; i < D_H; i += 256) h_out[b * D_H + i] = hbuf[i];
}

// ---------------------------------------------------------------------------
// Kernel 3: logits = h_last @ W_out^T + b_out.  64 x 16(pad, 10 valid) x 384
// fp32 WMMA GEMM; 1 block of 128 threads = 4 waves = 4 row tiles.
// Fake columns (o >= 10) read a clamped row of W_out and are masked to zero;
// all 32 lanes execute the WMMA with EXEC all-ones, stores are guarded after.
// ---------------------------------------------------------------------------
__global__ __launch_bounds__(128)
void logits_wmma(const float* __restrict__ h,      // (64,384)
                 const float* __restrict__ W_out,  // (10,384)
                 const float* __restrict__ b_out,  // (10)
                 float* __restrict__ out)          // (64,10)
{
    const int lane = threadIdx.x & 31;
    const int wave = threadIdx.x >> 5;   // row tile 0..3
    const int r0   = wave * 16;
    const int half = lane >> 4;
    const int l15  = lane & 15;

    const int   ocl   = (l15 < D_OUT) ? l15 : (D_OUT - 1); // in-bounds clamp
    const float omask = (l15 < D_OUT) ? 1.0f : 0.0f;

    v8f acc = {};
    const float* arow = h + (r0 + l15) * D_H;
    const float* wrow = W_out + ocl * D_H;
    for (int kk = 0; kk < D_H; kk += 4) {
        const int ka = kk + 2 * half;
        v2f a, bf;
        a.x  = arow[ka];          a.y  = arow[ka + 1];
        bf.x = wrow[ka] * omask;  bf.y = wrow[ka + 1] * omask;
        acc = __builtin_amdgcn_wmma_f32_16x16x4_f32(
                  false, a, false, bf, (short)0, acc, false, false);
    }
    if (l15 < D_OUT) {
        const float bias = b_out[l15];
        #pragma unroll
        for (int vr = 0; vr < 8; ++vr) {
            const int m = r0 + vr + 8 * half;
            out[m * D_OUT + l15] = acc[vr] + bias;
        }
    }
}

// ---------------------------------------------------------------------------
extern "C" void kernel_launch(void* const* d_in, const int* in_sizes, int n_in,
                              void* d_out, int out_size, void* d_ws, size_t ws_size,
                              hipStream_t stream) {
    const float* x     = (const float*)d_in[0];  // (48,64,64)
    const float* W_in  = (const float*)d_in[1];  // (384,64)
    const float* b_in  = (const float*)d_in[2];  // (384)
    const float* W     = (const float*)d_in[3];  // (384,384)
    const float* W_out = (const float*)d_in[4];  // (10,384)
    const float* b_out = (const float*)d_in[5];  // (10)
    float*       out   = (float*)d_out;          // (64,10)

    float* v_ws = (float*)d_ws;                         // 3072*384 f32
    float* h_ws = v_ws + (size_t)T_STEPS * BSZ * D_H;   // 64*384  f32

    // 1) hoisted input GEMM (WMMA): 4608 tiles, 4 waves/block
    v_gemm_wmma<<<1152, 128, 0, stream>>>(x, W_in, b_in, v_ws);

    // 2) sequential scan, M[b] resident in 240 KB of LDS per WGP
    const size_t lds_bytes =
        (size_t)(NUPPER * BLKELEM + D_H + D_H + 256) * sizeof(float); // ~250 KB
    plastic_scan<<<BSZ, 256, lds_bytes, stream>>>(v_ws, W, h_ws);

    // 3) output GEMM (WMMA)
    logits_wmma<<<1, 128, 0, stream>>>(h_ws, W_out, b_out, out);
}